// BezierDeformableAttention_50371376448061
// MI455X (gfx1250) — compile-verified
//
#include <hip/hip_runtime.h>
#include <hip/hip_bf16.h>

typedef float v2f __attribute__((ext_vector_type(2)));
typedef float v8f __attribute__((ext_vector_type(8)));

#define HF_   200
#define WF_   200
#define HWSZ  (HF_ * WF_)
#define BB    4
#define NQ    1024
#define DMODEL 256
#define NHEADS 8
#define NPTS  4
#define HDIM  32
#define KSAMP 10

// ---------------- WMMA f32 GEMM ----------------
// C[M,N] = A * Bw[K,N] + bias (+ resid).  A is generically strided:
// A[m,k] = A[m*ldaRow + k*ldaCol], so bev (C, H*W) feeds in without transpose.
// LDS strides of 36 floats keep ds_load_b64 8B-aligned and, for the half-wave
// fragment pattern (lanes 0-15 at +0, lanes 16-31 at +2), hit all 64 banks
// exactly once per fragment load (36*l mod 64 covers the multiples of 4).
#define BLK_M 128
#define BLK_N 64
#define KC    32
#define LDK   36   // padded K-stride for both LDS tiles

__global__ __launch_bounds__(256)
void wmma_gemm_f32(const float* __restrict__ A, const float* __restrict__ Bw,
                   const float* __restrict__ bias, const float* __restrict__ resid,
                   float* __restrict__ C, int M, int N, int K,
                   int ldaRow, int ldaCol, long long bsA, long long bsC)
{
    __shared__ __align__(16) float lA [BLK_M * LDK];  // [m][k]
    __shared__ __align__(16) float lBt[BLK_N * LDK];  // [n][k]  (transposed!)

    const int tid  = threadIdx.x;
    const int lane = tid & 31;
    const int wave = tid >> 5;      // 0..7 -> 16-row strip within 128-row tile
    const int l16  = lane & 15;
    const int hi   = lane >> 4;     // lane half selects K pair / M half

    const int nBlock = blockIdx.x * BLK_N;
    const int mBlock = blockIdx.y * BLK_M;
    const float* Ab = A + (size_t)blockIdx.z * bsA;
    float*       Cb = C + (size_t)blockIdx.z * bsC;

    v8f acc[4] = {};   // 4 N-subtiles of 16x16 f32

    for (int k0 = 0; k0 < K; k0 += KC) {
        // ---- stage A tile (BLK_M x KC) ----
        if (ldaCol == 1) {           // row-major A: coalesce along k
            for (int e = tid; e < BLK_M * KC; e += 256) {
                int r = e >> 5, c = e & 31;
                int gm = mBlock + r, gk = k0 + c;
                float v = 0.0f;
                if (gm < M && gk < K) v = Ab[(size_t)gm * ldaRow + gk];
                lA[r * LDK + c] = v;
            }
        } else {                     // column-strided A (bev): coalesce along m
            for (int e = tid; e < BLK_M * KC; e += 256) {
                int c = e >> 7, r = e & 127;
                int gm = mBlock + r, gk = k0 + c;
                float v = 0.0f;
                if (gm < M && gk < K) v = Ab[(size_t)gm * ldaRow + (size_t)gk * ldaCol];
                lA[r * LDK + c] = v;
            }
        }
        // ---- stage B tile (KC x BLK_N) transposed into lBt[n][k] ----
        for (int e = tid; e < KC * BLK_N; e += 256) {
            int r = e >> 6, c = e & 63;          // r = k (coalesced cols in global)
            int gk = k0 + r, gn = nBlock + c;
            float v = 0.0f;
            if (gk < K && gn < N) v = Bw[(size_t)gk * N + gn];
            lBt[c * LDK + r] = v;
        }
        // ---- prefetch next K-chunk while this one computes ----
        int kn = k0 + KC;
        if (kn < K) {
            {   // one A cacheline-ish probe per thread
                int e = tid << 4;                 // spread across the tile
                int r, c;
                if (ldaCol == 1) { r = e >> 5, c = e & 31; }
                else             { c = e >> 7, r = e & 127; }
                int gm = mBlock + r, gk = kn + c;
                if (gm < M && gk < K)
                    __builtin_prefetch(&Ab[(size_t)gm * ldaRow + (size_t)gk * ldaCol], 0, 1);
            }
            {   // B probe
                int e = tid << 3;
                int r = e >> 6, c = e & 63;
                int gk = kn + r, gn = nBlock + c;
                if (gk < K && gn < N)
                    __builtin_prefetch(&Bw[(size_t)gk * N + gn], 0, 1);
            }
        }
        __syncthreads();

        const int mBase = wave * 16;
        for (int kk = 0; kk < KC; kk += 4) {
            // A 16x4 frag: lanes 0-15 K={kk,kk+1}, lanes 16-31 K={kk+2,kk+3}
            v2f a = *(const v2f*)&lA[(mBase + l16) * LDK + kk + 2 * hi];
#pragma unroll
            for (int nb = 0; nb < 4; ++nb) {
                // B 4x16 frag from transposed tile: single aligned ds_load_b64
                v2f b = *(const v2f*)&lBt[(nb * 16 + l16) * LDK + kk + 2 * hi];
                acc[nb] = __builtin_amdgcn_wmma_f32_16x16x4_f32(
                    false, a, false, b, (short)0, acc[nb], false, false);
            }
        }
        __syncthreads();
    }

    // ---- epilogue: D layout = VGPR r -> row r (lanes 0-15) / r+8 (lanes 16-31)
    const int mOut = mBlock + wave * 16 + hi * 8;
#pragma unroll
    for (int nb = 0; nb < 4; ++nb) {
        int col = nBlock + nb * 16 + l16;
        if (col >= N) continue;
        float bv = bias ? bias[col] : 0.0f;
#pragma unroll
        for (int r = 0; r < 8; ++r) {
            int row = mOut + r;
            if (row < M) {
                float v = acc[nb][r] + bv;
                if (resid) v += resid[(size_t)row * N + col];
                Cb[(size_t)row * N + col] = v;
            }
        }
    }
}

// ---------------- Bezier reference-center ----------------
__global__ void bezier_refcenter(const float* __restrict__ ctrl,
                                 const float* __restrict__ pc_range,
                                 float* __restrict__ refc, int total)
{
    int i = blockIdx.x * blockDim.x + threadIdx.x;
    if (i >= total) return;
    const float lox = pc_range[0], loy = pc_range[1];
    const float spx = pc_range[3] - pc_range[0];
    const float spy = pc_range[4] - pc_range[1];
    const float* p = ctrl + (size_t)i * 8;
    float p0x = p[0], p0y = p[1], p1x = p[2], p1y = p[3];
    float p2x = p[4], p2y = p[5], p3x = p[6], p3y = p[7];
    float sx = 0.0f, sy = 0.0f;
#pragma unroll
    for (int k = 0; k < KSAMP; ++k) {
        float t = (float)k / (float)(KSAMP - 1);
        float u = 1.0f - t;
        float c0 = u * u * u, c1 = 3.0f * u * u * t, c2 = 3.0f * u * t * t, c3 = t * t * t;
        float x = c0 * p0x + c1 * p1x + c2 * p2x + c3 * p3x;
        float y = c0 * p0y + c1 * p1y + c2 * p2y + c3 * p3y;
        sx += fminf(fmaxf((x - lox) / spx, 0.01f), 0.99f);
        sy += fminf(fmaxf((y - loy) / spy, 0.01f), 0.99f);
    }
    refc[i * 2 + 0] = sx * (1.0f / KSAMP);
    refc[i * 2 + 1] = sy * (1.0f / KSAMP);
}

// ---------------- softmax + bilinear sample + head-weighted sum ----------------
// One block per (b,n).  value layout: (b, hw, h*32+d) so channel gathers coalesce.
__global__ __launch_bounds__(256)
void msda_sample(const float* __restrict__ value,
                 const float* __restrict__ refc,
                 const float* __restrict__ offs,   // (B*N, 64)
                 const float* __restrict__ attn,   // (B*N, 32) logits
                 float* __restrict__ outS)         // (B*N, 256)
{
    const int m = blockIdx.x;          // b*NQ + n
    const int b = m >> 10;             // NQ = 1024
    const int tid = threadIdx.x;

    __shared__ float sFx[NHEADS][NPTS], sFy[NHEADS][NPTS], sW[NHEADS][NPTS];

    if (tid < NHEADS * NPTS) {
        int h = tid >> 2, p = tid & 3;
        float rx = refc[m * 2 + 0], ry = refc[m * 2 + 1];
        float ox = offs[(size_t)m * 64 + (h * NPTS + p) * 2 + 0];
        float oy = offs[(size_t)m * 64 + (h * NPTS + p) * 2 + 1];
        // x = ((2*loc-1)+1)*0.5*W - 0.5 = ref*W + off - 0.5
        sFx[h][p] = rx * (float)WF_ + ox - 0.5f;
        sFy[h][p] = ry * (float)HF_ + oy - 0.5f;
    }
    if (tid < NHEADS) {
        float l[NPTS];
        float mx = -1e30f;
#pragma unroll
        for (int p = 0; p < NPTS; ++p) {
            l[p] = attn[(size_t)m * 32 + tid * NPTS + p];
            mx = fmaxf(mx, l[p]);
        }
        float s = 0.0f;
#pragma unroll
        for (int p = 0; p < NPTS; ++p) { l[p] = __expf(l[p] - mx); s += l[p]; }
        float inv = 1.0f / s;
#pragma unroll
        for (int p = 0; p < NPTS; ++p) sW[tid][p] = l[p] * inv;
    }
    __syncthreads();

    const int h = tid >> 5;            // HDIM = 32
    const float* vb = value + ((size_t)b * HWSZ) * DMODEL + tid; // tid == h*32+d
    float acc = 0.0f;
#pragma unroll
    for (int p = 0; p < NPTS; ++p) {
        float fx = sFx[h][p], fy = sFy[h][p];
        float fx0 = floorf(fx), fy0 = floorf(fy);
        int x0 = (int)fx0, y0 = (int)fy0;
        int x1 = x0 + 1,   y1 = y0 + 1;
        float wx = fx - fx0, wy = fy - fy0;
        bool vx0 = (x0 >= 0) & (x0 < WF_), vx1 = (x1 >= 0) & (x1 < WF_);
        bool vy0 = (y0 >= 0) & (y0 < HF_), vy1 = (y1 >= 0) & (y1 < HF_);
        float s = 0.0f;
        if (vy0 & vx0) s += (1.0f - wx) * (1.0f - wy) * vb[(size_t)(y0 * WF_ + x0) * DMODEL];
        if (vy0 & vx1) s += wx * (1.0f - wy)          * vb[(size_t)(y0 * WF_ + x1) * DMODEL];
        if (vy1 & vx0) s += (1.0f - wx) * wy          * vb[(size_t)(y1 * WF_ + x0) * DMODEL];
        if (vy1 & vx1) s += wx * wy                   * vb[(size_t)(y1 * WF_ + x1) * DMODEL];
        acc += sW[h][p] * s;
    }
    outS[(size_t)m * DMODEL + tid] = acc;
}

// ---------------- launcher ----------------
extern "C" void kernel_launch(void* const* d_in, const int* in_sizes, int n_in,
                              void* d_out, int out_size, void* d_ws, size_t ws_size,
                              hipStream_t stream)
{
    (void)in_sizes; (void)n_in; (void)out_size; (void)ws_size;
    const float* query_embed = (const float*)d_in[0];
    const float* ctrl        = (const float*)d_in[1];
    const float* bev         = (const float*)d_in[2];
    const float* pc_range    = (const float*)d_in[4];
    const float* W_q   = (const float*)d_in[5];  const float* b_q   = (const float*)d_in[6];
    const float* W_val = (const float*)d_in[7];  const float* b_val = (const float*)d_in[8];
    const float* W_off = (const float*)d_in[9];  const float* b_off = (const float*)d_in[10];
    const float* W_at  = (const float*)d_in[11]; const float* b_at  = (const float*)d_in[12];
    const float* W_do  = (const float*)d_in[13]; const float* b_do  = (const float*)d_in[14];
    const float* W_o   = (const float*)d_in[15]; const float* b_o   = (const float*)d_in[16];
    float* out = (float*)d_out;

    const int M  = BB * NQ;          // 4096
    char* ws = (char*)d_ws;
    size_t off = 0;
    auto wsAlloc = [&](size_t bytes) -> float* {
        float* p = (float*)(ws + off);
        off += (bytes + 255) & ~(size_t)255;
        return p;
    };
    float* refc  = wsAlloc((size_t)M * 2 * sizeof(float));
    float* query = wsAlloc((size_t)M * DMODEL * sizeof(float));
    float* offsB = wsAlloc((size_t)M * 64 * sizeof(float));
    float* attnB = wsAlloc((size_t)M * 32 * sizeof(float));
    float* outS  = wsAlloc((size_t)M * DMODEL * sizeof(float));
    float* msda  = wsAlloc((size_t)M * DMODEL * sizeof(float));
    float* value = wsAlloc((size_t)BB * HWSZ * DMODEL * sizeof(float)); // 164 MB

    bezier_refcenter<<<(M + 255) / 256, 256, 0, stream>>>(ctrl, pc_range, refc, M);

    // query = query_embed @ W_q + b_q
    wmma_gemm_f32<<<dim3(DMODEL / BLK_N, M / BLK_M, 1), 256, 0, stream>>>(
        query_embed, W_q, b_q, nullptr, query, M, DMODEL, DMODEL, DMODEL, 1, 0, 0);

    // value = bev^T @ W_val + b_val   (A[hw,c] = bev[b][c][hw]; grid.z = batch)
    wmma_gemm_f32<<<dim3(DMODEL / BLK_N, (HWSZ + BLK_M - 1) / BLK_M, BB), 256, 0, stream>>>(
        bev, W_val, b_val, nullptr, value, HWSZ, DMODEL, DMODEL,
        1, HWSZ, (long long)DMODEL * HWSZ, (long long)HWSZ * DMODEL);

    // offsets = query @ W_off + b_off   (N=64)
    wmma_gemm_f32<<<dim3(1, M / BLK_M, 1), 256, 0, stream>>>(
        query, W_off, b_off, nullptr, offsB, M, 64, DMODEL, DMODEL, 1, 0, 0);

    // attn logits = query @ W_attn + b_attn   (N=32, column-guarded)
    wmma_gemm_f32<<<dim3(1, M / BLK_M, 1), 256, 0, stream>>>(
        query, W_at, b_at, nullptr, attnB, M, 32, DMODEL, DMODEL, 1, 0, 0);

    // softmax + bilinear sampling + head accumulation
    msda_sample<<<M, 256, 0, stream>>>(value, refc, offsB, attnB, outS);

    // msda = outS @ W_do + b_do + query  (residual)
    wmma_gemm_f32<<<dim3(DMODEL / BLK_N, M / BLK_M, 1), 256, 0, stream>>>(
        outS, W_do, b_do, query, msda, M, DMODEL, DMODEL, DMODEL, 1, 0, 0);

    // result = msda @ W_out + b_out
    wmma_gemm_f32<<<dim3(DMODEL / BLK_N, M / BLK_M, 1), 256, 0, stream>>>(
        msda, W_o, b_o, nullptr, out, M, DMODEL, DMODEL, DMODEL, 1, 0, 0);
}